// LSTM_12781822672976
// MI455X (gfx1250) — compile-verified
//
#include <hip/hip_runtime.h>

// ---------------------------------------------------------------------------
// Fused LSTM scan for MI455X (gfx1250, wave32).
//
// Shapes: S=2048, B=4096, I=2, H=4 (4H=16 gates), O=1.
//
// Design (roofline-driven):
//   * Single fused kernel: read x (64MB), write y (32MB); no x_gates
//     materialization (saves ~1GB of HBM traffic vs the reference's
//     precomputed [S,B,16] f32 buffer). HBM floor ~4us; the real limiter is
//     the serial per-step latency chain, so everything below minimizes it.
//   * One wave owns 16 batch rows for the whole sequence; h,c live in VGPRs.
//   * Recurrent matvec per step:  gates.T[16x16] = W_hh[16x4] @ h.T[4x16]
//     == one V_WMMA_F32_16X16X4_F32 (M=gate, K=hidden, N=batch).
//     A (=W_hh) is loop-invariant and stays resident in 2 VGPRs; WMMA issues
//     on the matrix pipe and co-executes with the activation VALU work.
//   * D layout (ISA 7.12.2): VGPR r, lanes0-15 -> M=r (gates 0..7 = i,f),
//     lanes16-31 -> M=r+8 (gates 8..15 = g,o), N = batch = lane&15.
//   * Half-wave gate exchange via v_permlanex16_b32 (VALU pipe, ~2 cycles)
//     instead of ds_swizzle: removes the LDS round trip + s_wait_dscnt
//     ladder from the recurrence critical path. Both halves then hold all
//     16 gate values and compute identical c/h/y (no divergence; EXEC stays
//     all-1s as WMMA requires; duplicate stores write identical data).
//   * Activations on gfx1250's native v_tanh_f32 (sigmoid via tanh).
//   * global_prefetch_b8 pulls x 8 steps ahead of the serial chain.
// ---------------------------------------------------------------------------

typedef float v2f __attribute__((ext_vector_type(2)));
typedef float v8f __attribute__((ext_vector_type(8)));

#define LSTM_S 2048
#define LSTM_B 4096

__device__ __forceinline__ float dev_tanh(float x) {
#if __has_builtin(__builtin_amdgcn_tanhf)
  return __builtin_amdgcn_tanhf(x);
#elif __has_builtin(__builtin_amdgcn_tanh_f32)
  return __builtin_amdgcn_tanh_f32(x);
#else
  float r;
  // v_nop after the trans op: HW needs 1 independent op before result use
  // (ISA 7.4 multicycle co-execution hazards); inline asm is opaque to the
  // scheduler so provide the spacing ourselves.
  asm volatile("v_tanh_f32 %0, %1\n\tv_nop" : "=v"(r) : "v"(x));
  return r;
#endif
}

__device__ __forceinline__ float dev_sigmoid(float x) {
  // sigmoid(x) = 0.5 + 0.5*tanh(0.5*x)  -> 1 trans + 2 VALU
  return fmaf(0.5f, dev_tanh(0.5f * x), 0.5f);
}

// Cross-half-wave exchange (SWAPX16): each lane reads the mirrored lane of
// the other 16-lane row. v_permlanex16_b32 with identity selectors does this
// entirely in the VALU pipe (no LDS, no DScnt wait on the serial path).
__device__ __forceinline__ float xhalf(float v) {
#if __has_builtin(__builtin_amdgcn_permlanex16)
  return __int_as_float(__builtin_amdgcn_permlanex16(
      __float_as_int(v), __float_as_int(v),
      0x76543210u, 0xFEDCBA98u, false, false));
#else
  // Fallback: ds_swizzle group-of-32, xor=0x10, or=0, and=0x1f -> 0x401f
  return __int_as_float(__builtin_amdgcn_ds_swizzle(__float_as_int(v), 0x401F));
#endif
}

__global__ __launch_bounds__(128)
void lstm_fused_kernel(const float* __restrict__ x,      // [S,B,2]
                       const float* __restrict__ W_ih,   // [16,2]
                       const float* __restrict__ W_hh,   // [16,4]
                       const float* __restrict__ b_ih,   // [16]
                       const float* __restrict__ b_hh,   // [16]
                       const float* __restrict__ W_out,  // [1,4]
                       const float* __restrict__ b_out,  // [1]
                       float* __restrict__ y) {          // [S,B,1]
  const int tid   = blockIdx.x * blockDim.x + threadIdx.x;
  const int wave  = tid >> 5;
  const int lane  = tid & 31;
  const int ln    = lane & 15;
  const bool hi   = lane >= 16;
  const int batch = wave * 16 + ln;          // each wave owns 16 batch rows

  // ---- A operand: W_hh in the 16x4 f32 A layout (loop-invariant) ----------
  // lane<16:  M=lane,     VGPR0=K0, VGPR1=K1
  // lane>=16: M=lane-16,  VGPR0=K2, VGPR1=K3
  // W_hh row-major [16][4]: float2 index = M*2 + (hi?1:0)
  const v2f A = ((const v2f*)W_hh)[ln * 2 + (hi ? 1 : 0)];

  // ---- Per-lane input-gate constants for the C (x-gates) build ------------
  // C register r carries gate (r + hi*8) for batch = lane&15.
  float w0[8], w1[8], bb[8];
#pragma unroll
  for (int r = 0; r < 8; ++r) {
    const int g = r + (hi ? 8 : 0);
    w0[r] = W_ih[2 * g + 0];
    w1[r] = W_ih[2 * g + 1];
    bb[r] = b_ih[g] + b_hh[g];
  }
  const float wo0 = W_out[0], wo1 = W_out[1], wo2 = W_out[2], wo3 = W_out[3];
  const float bo  = b_out[0];

  // ---- Recurrent state (per lane: batch = wave*16 + (lane&15)) ------------
  float hh[4] = {0.f, 0.f, 0.f, 0.f};
  float cc[4] = {0.f, 0.f, 0.f, 0.f};

  const float* px = x + (size_t)batch * 2;   // x[0][batch][0]
  float*       py = y + batch;               // y[0][batch]

  for (int s = 0; s < LSTM_S; ++s) {
    // Prefetch x 8 timesteps ahead (global_prefetch_b8); the x stream is the
    // only off-chip read and is fully off the recurrence critical path.
    __builtin_prefetch(px + (size_t)8 * LSTM_B * 2, 0, 0);

    const float x0 = px[0];
    const float x1 = px[1];

    // C = x @ W_ih.T + b_ih + b_hh, directly in the 16x16 C/D layout
    // (compiler emits these as v_pk_fma_f32 pairs).
    v8f C;
#pragma unroll
    for (int r = 0; r < 8; ++r)
      C[r] = fmaf(w0[r], x0, fmaf(w1[r], x1, bb[r]));

    // B operand: h.T in 4x16 B layout, assumed to mirror the documented A
    // striping (VGPR0 = K0 | K2 across half-waves, VGPR1 = K1 | K3).
    // Both halves hold h0..h3, so this is just 2 selects.
    v2f Bm;
    Bm[0] = hi ? hh[2] : hh[0];
    Bm[1] = hi ? hh[3] : hh[1];

    // gates.T[16x16] = W_hh @ h.T + x_gates  (one matrix-pipe instruction)
    v8f D = __builtin_amdgcn_wmma_f32_16x16x4_f32(
        false, A, false, Bm, (short)0, C, false, false);

    // Exchange half-waves in the VALU pipe: lower half has i,f (gates 0..7),
    // upper has g,o (gates 8..15). After the exchange every lane sees all 16
    // gate values for its batch row; both halves compute identical c/h/y.
    float sw[8];
#pragma unroll
    for (int r = 0; r < 8; ++r) sw[r] = xhalf(D[r]);

    float yv = bo;
#pragma unroll
    for (int k = 0; k < 4; ++k) {
      const float iv = hi ? sw[k]     : D[k];       // gate i[k]
      const float fv = hi ? sw[4 + k] : D[4 + k];   // gate f[k]
      const float gv = hi ? D[k]      : sw[k];      // gate g[k]
      const float ov = hi ? D[4 + k]  : sw[4 + k];  // gate o[k]

      const float ia = dev_sigmoid(iv);
      const float fa = dev_sigmoid(fv);
      const float ga = dev_tanh(gv);
      const float oa = dev_sigmoid(ov);

      cc[k] = fmaf(fa, cc[k], ia * ga);
      hh[k] = oa * dev_tanh(cc[k]);

      const float wo = (k == 0) ? wo0 : (k == 1) ? wo1 : (k == 2) ? wo2 : wo3;
      yv = fmaf(fmaxf(hh[k], 0.f), wo, yv);         // fused ReLU + Linear head
    }

    // Lanes b and b+16 store the identical value to the identical address:
    // no EXEC manipulation needed (keeps EXEC all-1s for the next WMMA).
    *py = yv;
    py += LSTM_B;
    px += (size_t)LSTM_B * 2;
  }
}

extern "C" void kernel_launch(void* const* d_in, const int* in_sizes, int n_in,
                              void* d_out, int out_size, void* d_ws, size_t ws_size,
                              hipStream_t stream) {
  (void)in_sizes; (void)n_in; (void)out_size; (void)d_ws; (void)ws_size;
  const float* x     = (const float*)d_in[0];
  const float* W_ih  = (const float*)d_in[1];
  const float* W_hh  = (const float*)d_in[2];
  const float* b_ih  = (const float*)d_in[3];
  const float* b_hh  = (const float*)d_in[4];
  const float* W_out = (const float*)d_in[5];
  const float* b_out = (const float*)d_in[6];
  float* out = (float*)d_out;

  // One wave (32 lanes) per 16 batch rows -> 256 waves -> 8192 threads.
  const int threads = (LSTM_B / 16) * 32;
  const int block = 128;                 // 4 waves -> one per SIMD32 in a WGP
  const int grid = threads / block;      // 64 workgroups
  lstm_fused_kernel<<<grid, block, 0, stream>>>(x, W_ih, W_hh, b_ih, b_hh,
                                                W_out, b_out, out);
}